// HyperedgeMaxAggregator_73469710565693
// MI455X (gfx1250) — compile-verified
//
#include <hip/hip_runtime.h>

// Hyperedge segment-max aggregator for MI455X (gfx1250, wave32).
//   out[s, :] = max over members m with segment_ids[m]==s of emb[node_idx[m], :]
//   empty / non-finite -> 0  (matches jnp.where(isfinite, out, 0))
//
// One wave32 per segment; lane owns 4 of 128 features, so each member row is
// one fully-coalesced 512B wave read (global_load_b128 per lane). Member
// indices are double-buffered into LDS via async global->LDS DMA (ASYNCcnt),
// rows are software-prefetched 8 members ahead, and output is streamed with
// non-temporal stores so the 51MB table stays resident in the 192MB L2.

#define WPB   8            // waves per block
#define TPB   (WPB * 32)
#define DFEAT 128          // feature dim
#define VPL   4            // floats per lane = DFEAT / 32
#define CHUNK 32           // members staged per async chunk
#define PFD   8            // prefetch distance (members)

typedef float f32x4 __attribute__((ext_vector_type(4)));

#if __has_builtin(__builtin_amdgcn_global_load_async_to_lds_b32) && \
    __has_builtin(__builtin_amdgcn_s_wait_asynccnt)
#define USE_ASYNC_LDS 1
typedef __attribute__((address_space(1))) int* gas1_int;   // global int*
typedef __attribute__((address_space(3))) int* gas3_int;   // LDS int*
#else
#define USE_ASYNC_LDS 0
#endif

static __device__ __forceinline__ void async_wait_all() {
#if USE_ASYNC_LDS
    __builtin_amdgcn_s_wait_asynccnt(0);
    asm volatile("" ::: "memory");   // compiler barrier: LDS was written by async DMA
#endif
}

static __device__ __forceinline__ float finite_or_zero(float v) {
    // isfinite(v) ? v : 0   (NaN and +/-inf compare false)
    return (__builtin_fabsf(v) < __builtin_inff()) ? v : 0.0f;
}

__global__ __launch_bounds__(TPB) void hyperedge_segmax_kernel(
    const float* __restrict__ emb,       // [N_NODES, 128]
    const int*   __restrict__ node_idx,  // [FLAT]
    const int*   __restrict__ seg_ids,   // [FLAT], sorted ascending
    float*       __restrict__ out,       // [num_seg, 128]
    int flat, int num_seg)
{
    __shared__ int sidx[WPB][2][CHUNK];

    const int w    = threadIdx.x >> 5;
    const int lane = threadIdx.x & 31;
    const int s    = blockIdx.x * WPB + w;
    if (s >= num_seg) return;

    // --- segment extent via two lower_bound binary searches (wave-uniform) ---
    int lo = 0, hi = flat;
    while (lo < hi) {                       // lower_bound(seg_ids, s)
        int mid = (lo + hi) >> 1;
        if (seg_ids[mid] < s) lo = mid + 1; else hi = mid;
    }
    const int start = lo;
    hi = flat;
    while (lo < hi) {                       // lower_bound(seg_ids, s+1)
        int mid = (lo + hi) >> 1;
        if (seg_ids[mid] <= s) lo = mid + 1; else hi = mid;
    }
    const int end = lo;

    const float NEG_INF = -__builtin_inff();
    float a0 = NEG_INF, a1 = NEG_INF, a2 = NEG_INF, a3 = NEG_INF;

    const int nchunk = (end - start + CHUNK - 1) / CHUNK;

    // --- stage a chunk of member indices into LDS (async DMA when available) ---
    auto stage = [&](int m0, int b) {
        int cnt = end - m0; if (cnt > CHUNK) cnt = CHUNK;
        if (lane < cnt) {
#if USE_ASYNC_LDS
            __builtin_amdgcn_global_load_async_to_lds_b32(
                (gas1_int)(uintptr_t)(node_idx + m0 + lane),
                (gas3_int)(unsigned)(uintptr_t)&sidx[w][b][lane],
                /*offset=*/0, /*cpol=*/0);
#else
            sidx[w][b][lane] = node_idx[m0 + lane];
#endif
        }
    };

    int buf = 0;
    if (nchunk > 0) {
        stage(start, 0);
        async_wait_all();
    }

    for (int c = 0; c < nchunk; ++c) {
        const int m0 = start + c * CHUNK;
        int cnt = end - m0; if (cnt > CHUNK) cnt = CHUNK;

        // kick off async staging of the next chunk into the other buffer
        if (c + 1 < nchunk) stage(m0 + CHUNK, buf ^ 1);

        #pragma unroll 4
        for (int j = 0; j < cnt; ++j) {
            const int idx = sidx[w][buf][j];

            // prefetch the row PFD members ahead (each lane pulls its 16B slice)
            int pj = j + PFD; if (pj > cnt - 1) pj = cnt - 1;
            const int pidx = sidx[w][buf][pj];
            __builtin_prefetch(emb + (size_t)pidx * DFEAT + lane * VPL, 0, 1);

            const f32x4 v = *reinterpret_cast<const f32x4*>(
                emb + (size_t)idx * DFEAT + lane * VPL);
            a0 = fmaxf(a0, v.x);
            a1 = fmaxf(a1, v.y);
            a2 = fmaxf(a2, v.z);
            a3 = fmaxf(a3, v.w);
        }

        if (c + 1 < nchunk) async_wait_all();
        buf ^= 1;
    }

    f32x4 r;
    r.x = finite_or_zero(a0);
    r.y = finite_or_zero(a1);
    r.z = finite_or_zero(a2);
    r.w = finite_or_zero(a3);
    // streaming store: don't pollute L2 (keep the emb table resident)
    __builtin_nontemporal_store(
        r, reinterpret_cast<f32x4*>(out + (size_t)s * DFEAT + lane * VPL));
}

extern "C" void kernel_launch(void* const* d_in, const int* in_sizes, int n_in,
                              void* d_out, int out_size, void* d_ws, size_t ws_size,
                              hipStream_t stream) {
    (void)n_in; (void)d_ws; (void)ws_size;
    const float* emb      = (const float*)d_in[0];
    const int*   node_idx = (const int*)  d_in[1];
    const int*   seg_ids  = (const int*)  d_in[2];
    float*       out      = (float*)      d_out;

    const int flat    = in_sizes[1];        // total memberships
    const int num_seg = out_size / DFEAT;   // N_EDGES

    const int blocks = (num_seg + WPB - 1) / WPB;
    hipLaunchKernelGGL(hyperedge_segmax_kernel,
                       dim3(blocks), dim3(TPB), 0, stream,
                       emb, node_idx, seg_ids, out, flat, num_seg);
}